// SimpleGCN_55362128445711
// MI455X (gfx1250) — compile-verified
//
#include <hip/hip_runtime.h>
#include <hip/hip_bf16.h>
#include <math.h>

// ---------------- problem constants (match reference) ----------------
#define N_NODES   50000
#define N_EDGES   800000
#define NUM_GRAPHS 500
#define F_IN      128
#define HID       256
#define N_CLASSES 10

typedef __attribute__((ext_vector_type(2))) float v2f;
typedef __attribute__((ext_vector_type(8))) float v8f;

// =====================================================================
// f32 WMMA GEMM, LDS-staged B panel in k-pair-interleaved layout:
//   C[M x N] = A[M x K] * B[K x N] (+ bias[N])
// LDS layout: element (k, n) of the 64-wide B strip lives at float index
//   (k>>1)*128 + 2*n + (k&1)
// so the WMMA B fragment {B[k][n], B[k+1][n]} is ONE aligned ds_load_b64
// directly into the operand VGPR pair (no shuffle movs).
// Block = 8 waves; each wave computes TWO 16(M) x 64(N) tiles (8 v8f
// accumulators), so each K-step is: 2 global b64 (A) + 4 ds b64 (B)
// + 8 V_WMMA_F32_16X16X4_F32.
// Requires: M % 16 == 0, N % 64 == 0, K % 4 == 0. (50000 = 3125*16.)
// =====================================================================
__global__ __launch_bounds__(256)
void gemm_bias_wmma_f32(const float* __restrict__ A,
                        const float* __restrict__ B,
                        const float* __restrict__ bias,
                        float* __restrict__ C,
                        int M, int K, int N) {
  extern __shared__ float bsm[];              // K/2 x 128 interleaved B panel
  const int tid  = threadIdx.x;
  const int lane = tid & 31;
  const int wave = tid >> 5;
  const int Mt   = M >> 4;                    // 16-row tiles
  const int nbase = blockIdx.y << 6;

  // ---- stage B[:, nbase:nbase+64] into LDS, k-pair interleaved ----
  // one b128 store packs {B[2p][n], B[2p+1][n], B[2p][n+1], B[2p+1][n+1]}
  for (int i = tid; i < (K << 4); i += 256) { // (K/2 pair-rows) * 32 col-pairs
    const int p = i >> 5;
    const int g = i & 31;                     // column pair: n = 2g
    const float* r0 = B + (size_t)(2 * p) * N + nbase + 2 * g;
    const float* r1 = r0 + N;
    float4 v;
    v.x = r0[0]; v.y = r1[0]; v.z = r0[1]; v.w = r1[1];
    *(float4*)(bsm + p * 128 + 4 * g) = v;
  }
  __syncthreads();

  const int t0 = (blockIdx.x * 8 + wave) * 2; // this wave's two 16-row tiles
  const bool v0 = (t0 < Mt);
  const bool v1 = (t0 + 1 < Mt);
  const int t1 = v1 ? (t0 + 1) : t0;          // clamp (redundant compute, store guarded)

  if (v0) {                                   // wave-uniform: EXEC all-1 inside
    const int half = lane >> 4;               // 0: K rows {0,1}, 1: rows {2,3}
    const int l    = lane & 15;
    const int ncol = nbase + l;

    v8f acc0 = {}, acc1 = {}, acc2 = {}, acc3 = {};
    v8f acc4 = {}, acc5 = {}, acc6 = {}, acc7 = {};
    const float* aptr0 = A + (size_t)(t0 * 16 + l) * K + (half << 1);
    const float* aptr1 = A + (size_t)(t1 * 16 + l) * K + (half << 1);
    // lane's LDS base: pair-row advance of `half`, column l
    const float* bp = bsm + (half << 7) + (l << 1);

    for (int k0 = 0; k0 < K; k0 += 4) {
      v2f a0 = *(const v2f*)(aptr0 + k0);     // A tile 0 fragment (b64)
      v2f a1 = *(const v2f*)(aptr1 + k0);     // A tile 1 fragment (b64)
      const float* bk = bp + (k0 << 6);       // pair-row k0/2 (+half)
      v2f bA = *(const v2f*)(bk);             // cols ncol      : ds_load_b64
      v2f bB = *(const v2f*)(bk + 32);        // cols ncol + 16
      v2f bC = *(const v2f*)(bk + 64);        // cols ncol + 32
      v2f bD = *(const v2f*)(bk + 96);        // cols ncol + 48
      acc0 = __builtin_amdgcn_wmma_f32_16x16x4_f32(false, a0, false, bA, (short)0, acc0, false, false);
      acc1 = __builtin_amdgcn_wmma_f32_16x16x4_f32(false, a0, false, bB, (short)0, acc1, false, false);
      acc2 = __builtin_amdgcn_wmma_f32_16x16x4_f32(false, a0, false, bC, (short)0, acc2, false, false);
      acc3 = __builtin_amdgcn_wmma_f32_16x16x4_f32(false, a0, false, bD, (short)0, acc3, false, false);
      acc4 = __builtin_amdgcn_wmma_f32_16x16x4_f32(false, a1, false, bA, (short)0, acc4, false, false);
      acc5 = __builtin_amdgcn_wmma_f32_16x16x4_f32(false, a1, false, bB, (short)0, acc5, false, false);
      acc6 = __builtin_amdgcn_wmma_f32_16x16x4_f32(false, a1, false, bC, (short)0, acc6, false, false);
      acc7 = __builtin_amdgcn_wmma_f32_16x16x4_f32(false, a1, false, bD, (short)0, acc7, false, false);
    }

    float bv0 = 0.f, bv1 = 0.f, bv2 = 0.f, bv3 = 0.f;
    if (bias) {
      bv0 = bias[ncol];      bv1 = bias[ncol + 16];
      bv2 = bias[ncol + 32]; bv3 = bias[ncol + 48];
    }
    // C/D layout: VGPR r -> row tile*16 + r + 8*half, col = ncol (+16*j)
    #pragma unroll
    for (int r = 0; r < 8; ++r) {
      const int orow = t0 * 16 + r + (half << 3);
      float* crow = C + (size_t)orow * N + ncol;
      crow[0]  = acc0[r] + bv0;
      crow[16] = acc1[r] + bv1;
      crow[32] = acc2[r] + bv2;
      crow[48] = acc3[r] + bv3;
    }
    if (v1) {
      #pragma unroll
      for (int r = 0; r < 8; ++r) {
        const int orow = t1 * 16 + r + (half << 3);
        float* crow = C + (size_t)orow * N + ncol;
        crow[0]  = acc4[r] + bv0;
        crow[16] = acc5[r] + bv1;
        crow[32] = acc6[r] + bv2;
        crow[48] = acc7[r] + bv3;
      }
    }
  }
}

// =====================================================================
// Utility / graph kernels
// =====================================================================
__global__ void fill_f32(float* __restrict__ p, float v, size_t n) {
  size_t i = (size_t)blockIdx.x * blockDim.x + threadIdx.x;
  size_t stride = (size_t)gridDim.x * blockDim.x;
  for (; i < n; i += stride) p[i] = v;
}

// deg[dst] += w per edge (deg pre-filled with 1.0 for the self loop)
__global__ void edge_deg_kernel(const int* __restrict__ dst,
                                const float* __restrict__ ew,
                                float* __restrict__ deg, int n_edges) {
  int e = blockIdx.x * blockDim.x + threadIdx.x;
  if (e < n_edges) atomicAdd(&deg[dst[e]], ew[e]);
}

__global__ void dinv_kernel(const float* __restrict__ deg,
                            float* __restrict__ dinv, int n) {
  int i = blockIdx.x * blockDim.x + threadIdx.x;
  if (i < n) {
    float d = deg[i];
    dinv[i] = (d > 0.f) ? rsqrtf(fmaxf(d, 1e-12f)) : 0.f;
  }
}

// acc[dst] += hw[src] * (dinv[src]*w*dinv[dst]); 64 float4-threads per edge
__global__ __launch_bounds__(256)
void scatter_edges(const float* __restrict__ hw,
                   const int* __restrict__ src,
                   const int* __restrict__ dst,
                   const float* __restrict__ ew,
                   const float* __restrict__ dinv,
                   float* __restrict__ acc, int n_edges) {
  size_t gid = (size_t)blockIdx.x * blockDim.x + threadIdx.x;
  int e = (int)(gid >> 6);
  int c = (int)(gid & 63) << 2;           // 4 features per thread
  if (e >= n_edges) return;
  int s = src[e], d = dst[e];
  float coef = dinv[s] * ew[e] * dinv[d];
  const float4 hv = *(const float4*)(hw + (size_t)s * HID + c);
  float* o = acc + (size_t)d * HID + c;
  atomicAdd(o + 0, hv.x * coef);
  atomicAdd(o + 1, hv.y * coef);
  atomicAdd(o + 2, hv.z * coef);
  atomicAdd(o + 3, hv.w * coef);
}

// h = elu(acc + hw*dinv^2 (self loop) + bias)
__global__ __launch_bounds__(256)
void finalize_kernel(const float* __restrict__ acc,
                     const float* __restrict__ hw,
                     const float* __restrict__ dinv,
                     const float* __restrict__ bias,
                     float* __restrict__ h) {
  size_t gid = (size_t)blockIdx.x * blockDim.x + threadIdx.x;
  if (gid >= (size_t)N_NODES * HID) return;
  int n = (int)(gid >> 8);
  int f = (int)(gid & (HID - 1));
  float di = dinv[n];
  float v = acc[gid] + hw[gid] * di * di + bias[f];
  h[gid] = (v > 0.f) ? v : (expf(v) - 1.f);
}

__global__ __launch_bounds__(256)
void pool_sum_kernel(const float* __restrict__ h,
                     const int* __restrict__ batch,
                     float* __restrict__ sums) {
  size_t gid = (size_t)blockIdx.x * blockDim.x + threadIdx.x;
  if (gid >= (size_t)N_NODES * HID) return;
  int n = (int)(gid >> 8);
  int f = (int)(gid & (HID - 1));
  atomicAdd(&sums[(size_t)batch[n] * HID + f], h[gid]);
}

__global__ void pool_cnt_kernel(const int* __restrict__ batch,
                                float* __restrict__ cnts) {
  int n = blockIdx.x * blockDim.x + threadIdx.x;
  if (n < N_NODES) atomicAdd(&cnts[batch[n]], 1.0f);
}

// out[g,c] = (sums[g,:]/max(cnt,1)) @ lin1_w + lin1_b
__global__ void head_kernel(const float* __restrict__ sums,
                            const float* __restrict__ cnts,
                            const float* __restrict__ w,
                            const float* __restrict__ b,
                            float* __restrict__ out) {
  int tid = blockIdx.x * blockDim.x + threadIdx.x;
  if (tid >= NUM_GRAPHS * N_CLASSES) return;
  int g = tid / N_CLASSES;
  int c = tid % N_CLASSES;
  float inv = 1.0f / fmaxf(cnts[g], 1.0f);
  float s = b[c];
  const float* row = sums + (size_t)g * HID;
  for (int f = 0; f < HID; ++f) s += row[f] * inv * w[f * N_CLASSES + c];
  out[tid] = s;
}

// =====================================================================
// Host-side orchestration
// =====================================================================
static inline int ceil_div(size_t a, size_t b) { return (int)((a + b - 1) / b); }

extern "C" void kernel_launch(void* const* d_in, const int* in_sizes, int n_in,
                              void* d_out, int out_size, void* d_ws, size_t ws_size,
                              hipStream_t stream) {
  const float* x     = (const float*)d_in[0];
  const int*   ei    = (const int*)  d_in[1];   // [2, N_EDGES]: row 0 = src, row 1 = dst
  const float* ea    = (const float*)d_in[2];   // edge weights [N_EDGES,1]
  const int*   batch = (const int*)  d_in[4];
  const float* e1w   = (const float*)d_in[5];
  const float* e1b   = (const float*)d_in[6];
  const float* e2w   = (const float*)d_in[7];
  const float* e2b   = (const float*)d_in[8];
  const float* cws   = (const float*)d_in[9];   // [4, 256, 256]
  const float* cbs   = (const float*)d_in[10];  // [4, 256]
  const float* l1w   = (const float*)d_in[11];
  const float* l1b   = (const float*)d_in[12];
  float* out = (float*)d_out;

  const int* src = ei;
  const int* dst = ei + N_EDGES;

  // workspace carve-up (~154.5 MB of f32)
  float* h    = (float*)d_ws;                       // 50000*256
  float* hw   = h    + (size_t)N_NODES * HID;       // 50000*256
  float* acc  = hw   + (size_t)N_NODES * HID;       // 50000*256 (also encoder temp)
  float* deg  = acc  + (size_t)N_NODES * HID;       // 50000
  float* dinv = deg  + N_NODES;                     // 50000
  float* sums = dinv + N_NODES;                     // 500*256
  float* cnts = sums + (size_t)NUM_GRAPHS * HID;    // 500

  const size_t NH = (size_t)N_NODES * HID;
  const int MB = ceil_div(N_NODES / 16, 16);        // blocks of 8 waves x 2 tiles

  // ---- degree + normalization ----
  fill_f32<<<ceil_div(N_NODES, 256), 256, 0, stream>>>(deg, 1.0f, N_NODES); // self loops
  edge_deg_kernel<<<ceil_div(N_EDGES, 256), 256, 0, stream>>>(dst, ea, deg, N_EDGES);
  dinv_kernel<<<ceil_div(N_NODES, 256), 256, 0, stream>>>(deg, dinv, N_NODES);

  // ---- encoder: acc = x@enc1_w + b1 (N=128), h = acc@enc2_w + b2 (N=256) ----
  gemm_bias_wmma_f32<<<dim3(MB, F_IN / 64), 256, (size_t)F_IN * 64 * 4, stream>>>(
      x, e1w, e1b, acc, N_NODES, F_IN, F_IN);
  gemm_bias_wmma_f32<<<dim3(MB, HID / 64), 256, (size_t)F_IN * 64 * 4, stream>>>(
      acc, e2w, e2b, h, N_NODES, F_IN, HID);

  // ---- 4 GCN layers ----
  for (int layer = 0; layer < 4; ++layer) {
    const float* W = cws + (size_t)layer * HID * HID;
    const float* b = cbs + (size_t)layer * HID;
    gemm_bias_wmma_f32<<<dim3(MB, HID / 64), 256, (size_t)HID * 64 * 4, stream>>>(
        h, W, nullptr, hw, N_NODES, HID, HID);
    fill_f32<<<2048, 256, 0, stream>>>(acc, 0.0f, NH);
    scatter_edges<<<ceil_div((size_t)N_EDGES * 64, 256), 256, 0, stream>>>(
        hw, src, dst, ea, dinv, acc, N_EDGES);
    finalize_kernel<<<ceil_div(NH, 256), 256, 0, stream>>>(acc, hw, dinv, b, h);
  }

  // ---- mean pool + head ----
  fill_f32<<<ceil_div((size_t)NUM_GRAPHS * HID, 256), 256, 0, stream>>>(
      sums, 0.0f, (size_t)NUM_GRAPHS * HID);
  fill_f32<<<ceil_div(NUM_GRAPHS, 256), 256, 0, stream>>>(cnts, 0.0f, NUM_GRAPHS);
  pool_sum_kernel<<<ceil_div(NH, 256), 256, 0, stream>>>(h, batch, sums);
  pool_cnt_kernel<<<ceil_div(N_NODES, 256), 256, 0, stream>>>(batch, cnts);
  head_kernel<<<ceil_div(NUM_GRAPHS * N_CLASSES, 256), 256, 0, stream>>>(
      sums, cnts, l1w, l1b, out);
}